// HyposByFit_81733227643035
// MI455X (gfx1250) — compile-verified
//
#include <hip/hip_runtime.h>
#include <math.h>

typedef float v2f __attribute__((ext_vector_type(2)));
typedef float v8f __attribute__((ext_vector_type(8)));

#define NDEPTHS 16
#define NEG_LOG_095 0.051293294387550576f   // -ln(0.95)

// ---------------------------------------------------------------------------
// Pre-kernel: per-batch quadratic-fit weights  w_d = inv(XtX)[0,:] . [x^2,x,1]
// computed in double precision (64 values total, one block).
// ws layout (floats): [0 .. B*16)   w weights
//                     [B*16 + b]    cap_b  = 0.2*(dmax_b - dmin_b)
//                     [B*17 + b]    dmin_b
//                     [B*18 + b]    dmax_b
//                     [B*19]        up     = (max_b dmax - min_b dmin)/2
// ---------------------------------------------------------------------------
__global__ void precompute_kernel(const float* __restrict__ depth_range,
                                  const float* __restrict__ depth_hypos,
                                  float* __restrict__ ws, int B, int D) {
    int b = threadIdx.x;
    if (b < B) {
        double sx = 0.0, sx2 = 0.0, sx3 = 0.0, sx4 = 0.0;
        for (int d = 0; d < D; ++d) {
            double x  = (double)depth_hypos[b * D + d];
            double x2 = x * x;
            sx  += x;  sx2 += x2;  sx3 += x2 * x;  sx4 += x2 * x2;
        }
        // XtX = [[sx4,sx3,sx2],[sx3,sx2,sx],[sx2,sx,D]]; need row 0 of inverse.
        double a = sx4, bb = sx3, c = sx2, e = sx, f = (double)D;
        double C00 = c * f - e * e;
        double C10 = -(bb * f - c * e);
        double C20 = bb * e - c * c;
        double det = a * C00 + bb * C10 + c * C20;
        double i00 = C00 / det, i01 = C10 / det, i02 = C20 / det;
        for (int d = 0; d < D; ++d) {
            double x = (double)depth_hypos[b * D + d];
            ws[b * D + d] = (float)(i00 * x * x + i01 * x + i02);
        }
        float dmn = depth_range[2 * b], dmx = depth_range[2 * b + 1];
        ws[B * D + b]         = 0.2f * (dmx - dmn);
        ws[B * D + B + b]     = dmn;
        ws[B * D + 2 * B + b] = dmx;
    }
    if (threadIdx.x == 0) {
        float mx = depth_range[1], mn = depth_range[0];
        for (int i = 1; i < B; ++i) {
            mx = fmaxf(mx, depth_range[2 * i + 1]);
            mn = fminf(mn, depth_range[2 * i]);
        }
        ws[B * D + 3 * B] = 0.5f * (mx - mn);
    }
}

// ---------------------------------------------------------------------------
// Main kernel: one wave handles 32 consecutive pixels (two 16-pixel WMMA
// tiles). c0(pixel n) = sum_d w_d * log(prob[d,n]) via V_WMMA_F32_16X16X4_F32:
//   A (16x4)  = w_d broadcast to all 16 rows  -> every D row identical
//   B (4x16)  = z_d for pixel n in column n   (column == lane&15, certain)
// 4 chained WMMAs per tile cover K = 16 depths.
// ---------------------------------------------------------------------------
__global__ __launch_bounds__(256)
void hypos_wmma_kernel(const float* __restrict__ depth,
                       const float* __restrict__ prob,
                       const float* __restrict__ wsf,
                       float* __restrict__ out,
                       int B, int HW,
                       int capOff, int dminOff, int dmaxOff, int upOff) {
    const int lane = threadIdx.x & 31;
    const int wib  = threadIdx.x >> 5;
    const long long waveId  = (long long)blockIdx.x * (blockDim.x >> 5) + wib;
    const long long pixBase = waveId * 32;
    const long long total   = (long long)B * HW;
    if (pixBase >= total) return;                 // uniform per wave: EXEC stays full

    const int b = (int)(pixBase / HW);
    const long long rem       = pixBase - (long long)b * HW;       // offset in plane
    const long long planeBase = (long long)b * NDEPTHS * HW + rem + lane;

    // ---- stream in 16 depth-planes, 128B coalesced per plane, take log ----
    float z[NDEPTHS];
#pragma unroll
    for (int d = 0; d < NDEPTHS; ++d) {
        float p = prob[planeBase + (long long)d * HW];
        z[d] = logf(fmaxf(p, 1e-40f));
    }

    // ---- per-batch fit weights (uniform address -> scalar loads) ----
    const float* wv = wsf + b * NDEPTHS;

    const int half = lane >> 4;
    const int col  = lane & 15;
    const int idx0 = col * 4;            // ds_bpermute byte index: tile0 src lanes 0-15
    const int idx1 = (col + 16) * 4;     // tile1 src lanes 16-31

    v8f acc0 = {0.f, 0.f, 0.f, 0.f, 0.f, 0.f, 0.f, 0.f};
    v8f acc1 = {0.f, 0.f, 0.f, 0.f, 0.f, 0.f, 0.f, 0.f};

#pragma unroll
    for (int k = 0; k < 4; ++k) {
        // A fragment (16x4 f32): VGPR0 = K {0|2}, VGPR1 = K {1|3} per half-wave
        v2f a;
        a.x = wv[4 * k + 2 * half];
        a.y = wv[4 * k + 2 * half + 1];

        const int z0 = __float_as_int(z[4 * k + 0]);
        const int z1 = __float_as_int(z[4 * k + 1]);
        const int z2 = __float_as_int(z[4 * k + 2]);
        const int z3 = __float_as_int(z[4 * k + 3]);

        // ---- tile 0: gather z of pixels 0..15 into B layout ----
        {
            int t0 = __builtin_amdgcn_ds_bpermute(idx0, z0);
            int t2 = __builtin_amdgcn_ds_bpermute(idx0, z2);
            int u1 = __builtin_amdgcn_ds_bpermute(idx0, z1);
            int u3 = __builtin_amdgcn_ds_bpermute(idx0, z3);
            v2f bm;
            bm.x = __int_as_float(half ? t2 : t0);   // rows K {0|2}
            bm.y = __int_as_float(half ? u3 : u1);   // rows K {1|3}
            acc0 = __builtin_amdgcn_wmma_f32_16x16x4_f32(
                false, a, false, bm, (short)0, acc0, false, false);
        }
        // ---- tile 1: gather z of pixels 16..31 ----
        {
            int t0 = __builtin_amdgcn_ds_bpermute(idx1, z0);
            int t2 = __builtin_amdgcn_ds_bpermute(idx1, z2);
            int u1 = __builtin_amdgcn_ds_bpermute(idx1, z1);
            int u3 = __builtin_amdgcn_ds_bpermute(idx1, z3);
            v2f bm;
            bm.x = __int_as_float(half ? t2 : t0);
            bm.y = __int_as_float(half ? u3 : u1);
            acc1 = __builtin_amdgcn_wmma_f32_16x16x4_f32(
                false, a, false, bm, (short)0, acc1, false, false);
        }
    }

    // Every row of D is identical; element 0 of the accumulator in lane L is
    // c0 for column (L&15) of that tile -> lane L owns global pixel L.
    const float c0 = half ? acc1[0] : acc0[0];

    // ---- epilogue: depth_res + 16 hypotheses, 128B coalesced stores ----
    const float dpix = depth[pixBase + lane];
    float dres = sqrtf(NEG_LOG_095 / fabsf(c0));          // sqrt(-s*ln0.95), s=1/|c0|
    dres = fminf(fmaxf(dres, 1e-6f), wsf[upOff]);
    dres = fminf(dres, wsf[capOff + b]);
    const float dmn = wsf[dminOff + b];
    const float dmx = wsf[dmaxOff + b];
    const float interval = dres / (float)(NDEPTHS - 1);
    const float basev = dpix - 0.5f * dres;
#pragma unroll
    for (int d = 0; d < NDEPTHS; ++d) {
        float h = basev + (float)d * interval;
        h = dmn + fmaxf(h - dmn, 0.0f);   // replicate reference clamp exactly
        h = dmx + fminf(h - dmx, 0.0f);
        out[planeBase + (long long)d * HW] = h;
    }
}

// ---------------------------------------------------------------------------
extern "C" void kernel_launch(void* const* d_in, const int* in_sizes, int n_in,
                              void* d_out, int out_size, void* d_ws, size_t ws_size,
                              hipStream_t stream) {
    const float* depth       = (const float*)d_in[0];   // (B,H,W) f32
    const float* depth_range = (const float*)d_in[1];   // (B,2)   f32
    const float* prob        = (const float*)d_in[2];   // (B,D,H,W) f32
    const float* depth_hypos = (const float*)d_in[3];   // (B,D,1,1) f32

    const int B  = in_sizes[1] / 2;
    const int D  = in_sizes[3] / B;          // == NDEPTHS
    const int HW = in_sizes[0] / B;

    float* wsf = (float*)d_ws;
    precompute_kernel<<<1, 64, 0, stream>>>(depth_range, depth_hypos, wsf, B, D);

    const long long total  = (long long)B * HW;
    const long long waves  = (total + 31) / 32;
    const long long blocks = (waves + 7) / 8;           // 256 threads = 8 waves
    hypos_wmma_kernel<<<(int)blocks, 256, 0, stream>>>(
        depth, prob, wsf, (float*)d_out, B, HW,
        B * D, B * D + B, B * D + 2 * B, B * D + 3 * B);
}